// QGNN_25649544692292
// MI455X (gfx1250) — compile-verified
//
#include <hip/hip_runtime.h>
#include <hip/hip_bf16.h>

typedef __attribute__((ext_vector_type(2))) float v2f;
typedef __attribute__((ext_vector_type(8))) float v8f;

#define N_NODES 100000
#define N_EDGES 800000
#define F 64

// Padded LDS row strides: even (8B-aligned k-pairs) and stride%64 != 0
// (16 B-fragment lanes hit 16 distinct banks: 4n, 2n, 2n mod 64).
#define SW1_STRIDE 68    // W1 rows of 67
#define SW2_STRIDE 130   // W2 rows of 128
#define SWL_STRIDE 66    // Wl rows of 64

// D = A(16x4,f32) * B(4x16,f32) + C(16x16,f32), wave32.
__device__ inline v8f wmma4(v2f a, v2f b, v8f c) {
  return __builtin_amdgcn_wmma_f32_16x16x4_f32(
      /*neg_a=*/false, a, /*neg_b=*/false, b,
      /*c_mod=*/(short)0, c, /*reuse_a=*/false, /*reuse_b=*/false);
}

__global__ void k_clear(float* __restrict__ p, int n) {
  int i = blockIdx.x * blockDim.x + threadIdx.x;
  if (i < n) p[i] = 0.0f;
}

__global__ void k_deg(const int* __restrict__ dst, float* __restrict__ deg) {
  int i = blockIdx.x * blockDim.x + threadIdx.x;
  if (i < N_EDGES) atomicAdd(&deg[(unsigned)dst[i]], 1.0f);
}

__global__ void k_invdeg(float* __restrict__ deg) {
  int i = blockIdx.x * blockDim.x + threadIdx.x;
  if (i < N_NODES) deg[i] = 1.0f / fmaxf(deg[i], 1.0f);
}

__global__ void k_embed(const int* __restrict__ gt, const float* __restrict__ emb,
                        float* __restrict__ h) {
  int i = blockIdx.x * blockDim.x + threadIdx.x;
  if (i < N_NODES * F) {
    unsigned node = (unsigned)i >> 6, f = (unsigned)i & 63u;
    h[i] = emb[(unsigned)gt[node] * 64u + f];
  }
}

// Edge GEMM: tmp = leaky_relu([h[src], w] @ W1.T), atomically scattered to hN[dst].
// One wave = 16 edges x 64 outputs (4 N-tiles x 16 WMMA K=4). W1 staged in LDS.
__global__ void k_edge(const float* __restrict__ h,
                       const int* __restrict__ src, const int* __restrict__ dst,
                       const float* __restrict__ sidx, const float* __restrict__ didx,
                       const float* __restrict__ rev,
                       const float* __restrict__ W1,  // [64,67] row-major
                       float* __restrict__ hN) {
  __shared__ float sW[64 * SW1_STRIDE];
  for (int i = threadIdx.x; i < 64 * 67; i += blockDim.x) {
    int row = i / 67, col = i % 67;
    sW[row * SW1_STRIDE + col] = W1[i];
  }
  __syncthreads();

  int wave = (blockIdx.x * blockDim.x + threadIdx.x) >> 5;
  unsigned lane = threadIdx.x & 31u;
  unsigned half = lane >> 4;   // K sub-pair select for A/B frags
  unsigned l16  = lane & 15u;  // row (A) / column (B,C) index
  unsigned e0 = (unsigned)wave * 16u;
  if (e0 >= N_EDGES) return;  // wave-uniform: EXEC stays all-ones for WMMA

  // A-fragment: A[l16][k], k = 4*s + 2*half + {0,1}. 32-bit row offsets.
  unsigned arow_off = (unsigned)src[e0 + l16] * 64u;
  float areg[32];
#pragma unroll
  for (int s = 0; s < 16; ++s) {
    unsigned k = 4u * s + half * 2u;
    areg[2 * s]     = h[arow_off + k];
    areg[2 * s + 1] = h[arow_off + k + 1u];
  }
  // Epilogue operands for this lane's 8 C rows (M = r + 8*half), hoisted.
  unsigned doff[8];
  float se[8], de[8], re[8];
#pragma unroll
  for (int r = 0; r < 8; ++r) {
    unsigned e = e0 + (unsigned)r + half * 8u;
    doff[r] = (unsigned)dst[e] * 64u;
    se[r] = sidx[e]; de[r] = didx[e]; re[r] = rev[e];
  }

  for (int nt = 0; nt < 4; ++nt) {
    v8f c = {};
    unsigned n = (unsigned)nt * 16u + l16;
    const float* Wrow = sW + n * SW1_STRIDE;  // B[k][n] = W1[n][k], from LDS
#pragma unroll
    for (int s = 0; s < 16; ++s) {
      unsigned k = 4u * s + half * 2u;
      v2f a = {areg[2 * s], areg[2 * s + 1]};
      v2f b = {Wrow[k], Wrow[k + 1]};
      c = wmma4(a, b, c);
    }
    // Fold in the 3 edge-feature columns (K=64..66), activation, scatter.
    float w64 = Wrow[64], w65 = Wrow[65], w66 = Wrow[66];
#pragma unroll
    for (int r = 0; r < 8; ++r) {
      float v = c[r] + se[r] * w64 + de[r] * w65 + re[r] * w66;
      v = (v >= 0.0f) ? v : 0.01f * v;  // leaky_relu
      atomicAdd(&hN[doff[r] + n], v);   // 32-bit voffset + SGPR base (GVS)
    }
  }
}

// Node update: h_out = relu(concat(h, hN*inv_deg) @ W2.T + b2). K = 128. W2 in LDS.
__global__ void k_node_update(const float* __restrict__ h, const float* __restrict__ hN,
                              const float* __restrict__ inv_deg,
                              const float* __restrict__ W2,  // [64,128]
                              const float* __restrict__ b2,  // [64]
                              float* __restrict__ h_out) {
  __shared__ float sW[64 * SW2_STRIDE];
  for (int i = threadIdx.x; i < 64 * 128; i += blockDim.x) {
    int row = i >> 7, col = i & 127;
    sW[row * SW2_STRIDE + col] = W2[i];
  }
  __syncthreads();

  int wave = (blockIdx.x * blockDim.x + threadIdx.x) >> 5;
  unsigned lane = threadIdx.x & 31u;
  unsigned half = lane >> 4, l16 = lane & 15u;
  unsigned t0 = (unsigned)wave * 16u;
  if (t0 >= N_NODES) return;  // after barrier: safe for partial blocks

  unsigned node = t0 + l16;
  unsigned row_off = node * 64u;
  float inv = inv_deg[node];

  float areg[64];
#pragma unroll
  for (int s = 0; s < 16; ++s) {  // K = 0..63 from h
    unsigned k = 4u * s + half * 2u;
    areg[2 * s]     = h[row_off + k];
    areg[2 * s + 1] = h[row_off + k + 1u];
  }
#pragma unroll
  for (int s = 0; s < 16; ++s) {  // K = 64..127 from hN * inv_deg (mean mailbox)
    unsigned k = 4u * s + half * 2u;
    areg[32 + 2 * s] = hN[row_off + k] * inv;
    areg[33 + 2 * s] = hN[row_off + k + 1u] * inv;
  }

  for (int nt = 0; nt < 4; ++nt) {
    v8f c = {};
    unsigned n = (unsigned)nt * 16u + l16;
    const float* Wrow = sW + n * SW2_STRIDE;
#pragma unroll
    for (int s = 0; s < 32; ++s) {
      unsigned k = 4u * s + half * 2u;
      v2f a = {areg[2 * s], areg[2 * s + 1]};
      v2f b = {Wrow[k], Wrow[k + 1]};
      c = wmma4(a, b, c);
    }
    float bias = b2[n];
#pragma unroll
    for (int r = 0; r < 8; ++r) {
      float v = fmaxf(c[r] + bias, 0.0f);
      h_out[(t0 + (unsigned)r + half * 8u) * 64u + n] = v;
    }
  }
}

// Dense layer: out = act(h @ W.T + b), K = 64, NOUT in {64, 32}. W in LDS.
template <int NOUT, bool RELU>
__global__ void k_linear(const float* __restrict__ h, const float* __restrict__ W,
                         const float* __restrict__ b, float* __restrict__ out) {
  __shared__ float sW[NOUT * SWL_STRIDE];
  for (int i = threadIdx.x; i < NOUT * 64; i += blockDim.x) {
    int row = i >> 6, col = i & 63;
    sW[row * SWL_STRIDE + col] = W[i];
  }
  __syncthreads();

  int wave = (blockIdx.x * blockDim.x + threadIdx.x) >> 5;
  unsigned lane = threadIdx.x & 31u;
  unsigned half = lane >> 4, l16 = lane & 15u;
  unsigned t0 = (unsigned)wave * 16u;
  if (t0 >= N_NODES) return;

  unsigned row_off = (t0 + l16) * 64u;
  float areg[32];
#pragma unroll
  for (int s = 0; s < 16; ++s) {
    unsigned k = 4u * s + half * 2u;
    areg[2 * s]     = h[row_off + k];
    areg[2 * s + 1] = h[row_off + k + 1u];
  }

  for (int nt = 0; nt < NOUT / 16; ++nt) {
    v8f c = {};
    unsigned n = (unsigned)nt * 16u + l16;
    const float* Wrow = sW + n * SWL_STRIDE;
#pragma unroll
    for (int s = 0; s < 16; ++s) {
      unsigned k = 4u * s + half * 2u;
      v2f a = {areg[2 * s], areg[2 * s + 1]};
      v2f bv = {Wrow[k], Wrow[k + 1]};
      c = wmma4(a, bv, c);
    }
    float bias = b[n];
#pragma unroll
    for (int r = 0; r < 8; ++r) {
      float v = c[r] + bias;
      if (RELU) v = fmaxf(v, 0.0f);
      out[(t0 + (unsigned)r + half * 8u) * (unsigned)NOUT + n] = v;
    }
  }
}

extern "C" void kernel_launch(void* const* d_in, const int* in_sizes, int n_in,
                              void* d_out, int out_size, void* d_ws, size_t ws_size,
                              hipStream_t stream) {
  (void)in_sizes; (void)n_in; (void)out_size; (void)ws_size;
  const int*   gate_type = (const int*)d_in[0];
  const int*   src       = (const int*)d_in[1];
  const int*   dst       = (const int*)d_in[2];
  const float* src_idx   = (const float*)d_in[3];
  const float* dst_idx   = (const float*)d_in[4];
  const float* rev       = (const float*)d_in[5];
  const float* emb       = (const float*)d_in[6];
  const float* W1        = (const float*)d_in[7];   // [5,64,67]
  const float* W2        = (const float*)d_in[8];   // [5,64,128]
  const float* b2        = (const float*)d_in[9];   // [5,64]
  const float* Wl1       = (const float*)d_in[10];  // [64,64]
  const float* bl1       = (const float*)d_in[11];
  const float* Wl2       = (const float*)d_in[12];  // [32,64]
  const float* bl2       = (const float*)d_in[13];

  float* ws      = (float*)d_ws;
  float* inv_deg = ws;                    // N
  float* hA      = inv_deg + N_NODES;     // N*64
  float* hB      = hA + (size_t)N_NODES * F;
  float* hN      = hB + (size_t)N_NODES * F;

  const int T = 256;
  k_clear<<<(N_NODES + T - 1) / T, T, 0, stream>>>(inv_deg, N_NODES);
  k_deg<<<(N_EDGES + T - 1) / T, T, 0, stream>>>(dst, inv_deg);
  k_invdeg<<<(N_NODES + T - 1) / T, T, 0, stream>>>(inv_deg);
  k_embed<<<(N_NODES * F + T - 1) / T, T, 0, stream>>>(gate_type, emb, hA);

  int edge_waves  = N_EDGES / 16;               // 50000 (divides exactly)
  int edge_blocks = (edge_waves + 7) / 8;       // 8 waves / 256-thread block
  int node_waves  = (N_NODES + 15) / 16;        // 6250 (divides exactly)
  int node_blocks = (node_waves + 7) / 8;

  float* hcur = hA;
  float* hnext = hB;
  for (int i = 0; i < 5; ++i) {
    k_clear<<<(N_NODES * F + T - 1) / T, T, 0, stream>>>(hN, N_NODES * F);
    k_edge<<<edge_blocks, T, 0, stream>>>(hcur, src, dst, src_idx, dst_idx, rev,
                                          W1 + (size_t)i * 64 * 67, hN);
    k_node_update<<<node_blocks, T, 0, stream>>>(hcur, hN, inv_deg,
                                                 W2 + (size_t)i * 64 * 128,
                                                 b2 + (size_t)i * 64, hnext);
    float* t = hcur; hcur = hnext; hnext = t;
  }
  k_linear<64, true><<<node_blocks, T, 0, stream>>>(hcur, Wl1, bl1, hnext);
  k_linear<32, false><<<node_blocks, T, 0, stream>>>(hnext, Wl2, bl2, (float*)d_out);
}